// GNNModel_65541200937586
// MI455X (gfx1250) — compile-verified
//
#include <hip/hip_runtime.h>

// ---------------------------------------------------------------------------
// GNN pipeline for MI455X (gfx1250, wave32).
// deg scatter (async-LDS double-buffered) -> dinv + emb@w_conv ->
// weighted message scatter -> act -> WMMA f32 MLP (LDS-padded weights,
// LDS micro-transpose for layer 2) -> column-wise log_softmax (2-stage LSE).
// ---------------------------------------------------------------------------

typedef __attribute__((ext_vector_type(2))) float v2f;
typedef __attribute__((ext_vector_type(8))) float v8f;

__device__ __forceinline__ float lrelu(float v) { return v > 0.0f ? v : 0.01f * v; }

// ---------------- 1. zero scratch (deg + agg) ----------------
__global__ void k_zero(float* __restrict__ p, int n) {
  int i = blockIdx.x * blockDim.x + threadIdx.x;
  int stride = gridDim.x * blockDim.x;
  for (; i < n; i += stride) p[i] = 0.0f;
}

// ---------------- 2. degree accumulation ----------------
// Double-buffered GLOBAL_LOAD_ASYNC_TO_LDS staging of the (col, weight) edge
// streams; async loads complete in order, so waiting asynccnt<=2 after issuing
// the next chunk's 2 loads guarantees the current chunk is resident.
#define CHUNK 256
__global__ void k_deg(const int* __restrict__ col, const float* __restrict__ ew,
                      float* __restrict__ deg, int E, int nChunks) {
  __shared__ int   sCol[2][CHUNK];
  __shared__ float sW[2][CHUNK];
  const int tid = threadIdx.x;
  if ((int)blockIdx.x >= nChunks) return;

  auto issue = [&](int chunk, int buf) {
    int e = chunk * CHUNK + tid;
    int ec = e < E ? e : (E - 1);  // clamp: keep EXEC full, harmless load
    unsigned ldsC = (unsigned)(size_t)&sCol[buf][tid];
    unsigned ldsW = (unsigned)(size_t)&sW[buf][tid];
    const int*   gc = col + ec;
    const float* gw = ew + ec;
    asm volatile("global_load_async_to_lds_b32 %0, %1, off"
                 :: "v"(ldsC), "v"(gc) : "memory");
    asm volatile("global_load_async_to_lds_b32 %0, %1, off"
                 :: "v"(ldsW), "v"(gw) : "memory");
  };

  issue(blockIdx.x, 0);
  int buf = 0;
  for (int chunk = blockIdx.x; chunk < nChunks; chunk += gridDim.x) {
    int next = chunk + gridDim.x;
    if (next < nChunks) {
      issue(next, buf ^ 1);
      asm volatile("s_wait_asynccnt 0x2" ::: "memory");  // prev chunk done
    } else {
      asm volatile("s_wait_asynccnt 0x0" ::: "memory");
    }
    int e = chunk * CHUNK + tid;
    if (e < E) atomicAdd(&deg[sCol[buf][tid]], sW[buf][tid]);
    buf ^= 1;
  }
}

// ---------------- 3. dinv + x = emb @ w_conv ----------------
__global__ void k_dinv_x(float* __restrict__ deg /*in: deg, out: dinv*/,
                         const float* __restrict__ emb,
                         const float* __restrict__ wc,  // [3,3] row-major
                         float* __restrict__ x, int N) {
  int i = blockIdx.x * blockDim.x + threadIdx.x;
  if (i >= N) return;
  float d = deg[i];
  deg[i] = d > 0.0f ? rsqrtf(d) : 0.0f;
  float e0 = emb[i * 3 + 0], e1 = emb[i * 3 + 1], e2 = emb[i * 3 + 2];
#pragma unroll
  for (int j = 0; j < 3; ++j)
    x[i * 3 + j] = e0 * wc[0 * 3 + j] + e1 * wc[1 * 3 + j] + e2 * wc[2 * 3 + j];
}

// ---------------- 4. message scatter ----------------
__global__ void k_scatter(const int* __restrict__ row, const int* __restrict__ col,
                          const float* __restrict__ ew, const float* __restrict__ dinv,
                          const float* __restrict__ x, float* __restrict__ agg, int E) {
  int e = blockIdx.x * blockDim.x + threadIdx.x;
  if (e >= E) return;
  __builtin_prefetch(row + e + 4096, 0, 0);
  __builtin_prefetch(col + e + 4096, 0, 0);
  __builtin_prefetch(ew  + e + 4096, 0, 0);
  int r = row[e], c = col[e];
  float nrm = dinv[r] * ew[e] * dinv[c];
  float b0 = x[r * 3 + 0], b1v = x[r * 3 + 1], b2 = x[r * 3 + 2];
  atomicAdd(&agg[c * 3 + 0], nrm * b0);
  atomicAdd(&agg[c * 3 + 1], nrm * b1v);
  atomicAdd(&agg[c * 3 + 2], nrm * b2);
}

// ---------------- 5. bias + leaky_relu ----------------
__global__ void k_act(float* __restrict__ agg, const float* __restrict__ bc, int N3) {
  int i = blockIdx.x * blockDim.x + threadIdx.x;
  if (i >= N3) return;
  agg[i] = lrelu(agg[i] + bc[i % 3]);
}

// ---------------- 6. MLP with f32 WMMA (16 rows / wave) ----------------
// layer1: h[16,6] @ w1[6,6] via two V_WMMA_F32_16X16X4_F32 (K padded 6->8,
// N padded 6->16; padding staged as zeros in LDS so everything is branchless).
// layer2 (6->3): per-wave LDS micro-transpose of the WMMA accumulator
// (lane holds hidden unit n = lane%16 for 8 rows), then lanes 0-15 each
// finish one batch row with 18 FMAs. No cross-lane permutes needed.
__global__ void k_mlp(const float* __restrict__ conv,  // [N,3] activated conv output
                      const int* __restrict__ home, const int* __restrict__ away,
                      const float* __restrict__ w1, const float* __restrict__ b1,
                      const float* __restrict__ w3, const float* __restrict__ b3,
                      float* __restrict__ out, int B) {
  __shared__ float w1p[8][16];    // zero-padded [K=8][N=16]
  __shared__ float b1p[16];
  __shared__ float w3p[8][4];     // zero-padded [K=8][3 (+1 pad)]
  __shared__ float t[8][16][17];  // per-wave transpose tile [wave][row][hidden]
  const int tid = threadIdx.x;
  if (tid < 128) {
    int k = tid >> 4, n = tid & 15;
    w1p[k][n] = (k < 6 && n < 6) ? w1[k * 6 + n] : 0.0f;
  }
  if (tid < 16) b1p[tid] = (tid < 6) ? b1[tid] : 0.0f;
  if (tid < 24) {
    int n = tid / 3, c = tid % 3;
    w3p[n][c] = (n < 6) ? w3[n * 3 + c] : 0.0f;
  }
  __syncthreads();

  const int lane    = tid & 31;
  const int waveId  = tid >> 5;
  const int rowBase = (blockIdx.x * (blockDim.x >> 5) + waveId) * 16;
  const int m    = lane & 15;       // batch row (A) / hidden unit n (B,C,D)
  const int half = lane >> 4;       // K-half selector for A/B fragments
  const bool hiHalf = (half != 0);

  // Gather this lane's batch row features (clamped; WMMA needs full EXEC).
  int rIdx = rowBase + m;
  int rClamp = rIdx < B ? rIdx : (B - 1);
  int hi = home[rClamp], ai = away[rClamp];
  float h0 = conv[hi * 3 + 0], h1 = conv[hi * 3 + 1], h2 = conv[hi * 3 + 2];
  float h3 = conv[ai * 3 + 0], h4 = conv[ai * 3 + 1], h5 = conv[ai * 3 + 2];

  v8f acc = {};
  v2f a, bfrag;
  // chunk 0: K = 0..3  (lane pair k = 2*half, 2*half+1)
  a.x = hiHalf ? h2 : h0;
  a.y = hiHalf ? h3 : h1;
  bfrag.x = w1p[2 * half + 0][m];
  bfrag.y = w1p[2 * half + 1][m];
  acc = __builtin_amdgcn_wmma_f32_16x16x4_f32(false, a, false, bfrag,
                                              (short)0, acc, false, false);
  // chunk 1: K = 4..7  (k = 4+2*half; k>=6 rows are zero-padded in LDS)
  a.x = hiHalf ? 0.0f : h4;
  a.y = hiHalf ? 0.0f : h5;
  bfrag.x = w1p[4 + 2 * half + 0][m];
  bfrag.y = w1p[4 + 2 * half + 1][m];
  acc = __builtin_amdgcn_wmma_f32_16x16x4_f32(false, a, false, bfrag,
                                              (short)0, acc, false, false);

  // Lane holds column n = m of the 16x16 result, rows M = r + 8*half.
  // Padded weights force hv == 0 for n >= 6 -> branchless everywhere.
  const int n = m;
  float bias1 = b1p[n];
#pragma unroll
  for (int r = 0; r < 8; ++r) {
    t[waveId][r + 8 * half][n] = lrelu(acc[r] + bias1);
  }
  __syncthreads();

  // Layer 2: lanes 0-15 each finish one batch row (6->3 with bias + lrelu).
  if (lane < 16) {
    int mg = rowBase + lane;
    if (mg < B) {
      float o0 = b3[0], o1 = b3[1], o2 = b3[2];
#pragma unroll
      for (int nn = 0; nn < 6; ++nn) {
        float hv = t[waveId][lane][nn];
        o0 += hv * w3p[nn][0];
        o1 += hv * w3p[nn][1];
        o2 += hv * w3p[nn][2];
      }
      out[mg * 3 + 0] = lrelu(o0);
      out[mg * 3 + 1] = lrelu(o1);
      out[mg * 3 + 2] = lrelu(o2);
    }
  }
}

// ---------------- 7. per-block online (max,sum) per column ----------------
#define RBLOCKS 512
__global__ void k_lse_partial(const float* __restrict__ logits,
                              float* __restrict__ pmax, float* __restrict__ psum, int B) {
  __shared__ float sm[3][256];
  __shared__ float ss[3][256];
  float m[3] = {-INFINITY, -INFINITY, -INFINITY};
  float s[3] = {0.0f, 0.0f, 0.0f};
  for (int i = blockIdx.x * blockDim.x + threadIdx.x; i < B; i += gridDim.x * blockDim.x) {
#pragma unroll
    for (int c = 0; c < 3; ++c) {
      float v = logits[i * 3 + c];
      if (v > m[c]) { s[c] = s[c] * __expf(m[c] - v) + 1.0f; m[c] = v; }
      else          { s[c] += __expf(v - m[c]); }
    }
  }
#pragma unroll
  for (int c = 0; c < 3; ++c) { sm[c][threadIdx.x] = m[c]; ss[c][threadIdx.x] = s[c]; }
  __syncthreads();
  for (int off = 128; off > 0; off >>= 1) {
    if (threadIdx.x < off) {
#pragma unroll
      for (int c = 0; c < 3; ++c) {
        float m1 = sm[c][threadIdx.x], s1 = ss[c][threadIdx.x];
        float m2 = sm[c][threadIdx.x + off], s2 = ss[c][threadIdx.x + off];
        if (s2 > 0.0f) {
          if (s1 > 0.0f) {
            float nm = m1 > m2 ? m1 : m2;
            s1 = s1 * __expf(m1 - nm) + s2 * __expf(m2 - nm);
            m1 = nm;
          } else { m1 = m2; s1 = s2; }
        }
        sm[c][threadIdx.x] = m1; ss[c][threadIdx.x] = s1;
      }
    }
    __syncthreads();
  }
  if (threadIdx.x < 3) {
    pmax[blockIdx.x * 3 + threadIdx.x] = sm[threadIdx.x][0];
    psum[blockIdx.x * 3 + threadIdx.x] = ss[threadIdx.x][0];
  }
}

// ---------------- 8. sequential combine -> L[c] = m + log(s) ----------------
__global__ void k_lse_combine(const float* __restrict__ pmax, const float* __restrict__ psum,
                              float* __restrict__ L, int nblocks) {
  int c = threadIdx.x;
  if (c >= 3) return;
  float m = -INFINITY, s = 0.0f;
  for (int b = 0; b < nblocks; ++b) {
    float pm = pmax[b * 3 + c], ps = psum[b * 3 + c];
    if (ps > 0.0f) {
      if (s > 0.0f) {
        float nm = m > pm ? m : pm;
        s = s * __expf(m - nm) + ps * __expf(pm - nm);
        m = nm;
      } else { m = pm; s = ps; }
    }
  }
  L[c] = m + logf(s);
}

// ---------------- 9. subtract logsumexp ----------------
__global__ void k_sub(float* __restrict__ out, const float* __restrict__ L, int total) {
  int i = blockIdx.x * blockDim.x + threadIdx.x;
  if (i >= total) return;
  out[i] -= L[i % 3];
}

// ---------------------------------------------------------------------------
extern "C" void kernel_launch(void* const* d_in, const int* in_sizes, int n_in,
                              void* d_out, int out_size, void* d_ws, size_t ws_size,
                              hipStream_t stream) {
  const int*   edge_index = (const int*)d_in[0];    // [2,E]
  const float* edge_w     = (const float*)d_in[1];  // [E]
  const int*   home       = (const int*)d_in[2];    // [B]
  const int*   away       = (const int*)d_in[3];    // [B]
  const float* emb        = (const float*)d_in[4];  // [N,3]
  const float* w_conv     = (const float*)d_in[5];  // [3,3]
  const float* b_conv     = (const float*)d_in[6];  // [3]
  const float* w1         = (const float*)d_in[7];  // [6,6]
  const float* b1         = (const float*)d_in[8];  // [6]
  const float* w3         = (const float*)d_in[9];  // [6,3]
  const float* b3         = (const float*)d_in[10]; // [3]

  const int E = in_sizes[1];
  const int B = in_sizes[2];
  const int N = in_sizes[4] / 3;

  const int* row = edge_index;      // edge_index[0]
  const int* col = edge_index + E;  // edge_index[1]

  // Workspace layout (floats)
  float* ws    = (float*)d_ws;
  float* deg   = ws;                 // N      (becomes dinv in-place)
  float* x     = deg + N;            // 3N     (emb @ w_conv)
  float* agg   = x + 3 * N;          // 3N     (conv output, in-place activated)
  float* pmax  = agg + 3 * N;        // RBLOCKS*3
  float* psum  = pmax + RBLOCKS * 3; // RBLOCKS*3
  float* Lse   = psum + RBLOCKS * 3; // 3

  const int T = 256;

  // 1. zero deg (N) and agg (3N)
  k_zero<<<2048, T, 0, stream>>>(deg, N);
  k_zero<<<2048, T, 0, stream>>>(agg, 3 * N);

  // 2. degree (async-LDS double-buffered edge streaming)
  int nChunks = (E + CHUNK - 1) / CHUNK;
  int degBlocks = nChunks < 4096 ? nChunks : 4096;
  k_deg<<<degBlocks, CHUNK, 0, stream>>>(col, edge_w, deg, E, nChunks);

  // 3. dinv + feature transform
  k_dinv_x<<<(N + T - 1) / T, T, 0, stream>>>(deg, emb, w_conv, x, N);

  // 4. weighted message scatter
  k_scatter<<<(E + T - 1) / T, T, 0, stream>>>(row, col, edge_w, deg, x, agg, E);

  // 5. bias + activation
  k_act<<<(3 * N + T - 1) / T, T, 0, stream>>>(agg, b_conv, 3 * N);

  // 6. WMMA MLP -> logits in d_out
  int rowsPerBlock = (T / 32) * 16;  // 128
  k_mlp<<<(B + rowsPerBlock - 1) / rowsPerBlock, T, 0, stream>>>(
      agg, home, away, w1, b1, w3, b3, (float*)d_out, B);

  // 7-9. log_softmax over axis=0
  k_lse_partial<<<RBLOCKS, T, 0, stream>>>((const float*)d_out, pmax, psum, B);
  k_lse_combine<<<1, 32, 0, stream>>>(pmax, psum, Lse, RBLOCKS);
  k_sub<<<(3 * B + T - 1) / T, T, 0, stream>>>((float*)d_out, Lse, 3 * B);
}